// HybridGNN_12317966205310
// MI455X (gfx1250) — compile-verified
//
#include <hip/hip_runtime.h>

// ---------------------------------------------------------------------------
// HybridGNN (3-layer GCN + global MLP head) for MI455X (gfx1250, wave32).
//
// Roofline: GEMMs are ~2.9 GFLOP (trivial; done with v_wmma_f32_16x16x4_f32).
// The aggregation (1.6M edges x 64 feats x 3 layers) is the hot loop; h2 is
// 25.6MB and lives in the 192MB L2, so it is an L2 gather + fp32-atomic
// problem. One wave per edge: 32 lanes x float2 = one 256B contiguous row
// read, 64 native global_atomic_add_f32 (scope DEV) into the destination row.
// ---------------------------------------------------------------------------

#define NNODES 100000
#define NEDGES 1600000
#define FNODE  32
#define FGLOB  16
#define HDIM   64
#define OUTDIM 64

typedef float v2f __attribute__((ext_vector_type(2)));
typedef float v8f __attribute__((ext_vector_type(8)));

// Native fp32 global atomic add (no return) at device scope. Inline asm so we
// are guaranteed GLOBAL_ATOMIC_ADD_F32 (not a CAS loop). Tracked by STOREcnt;
// S_ENDPGM does an implicit wait-idle, and "memory" clobber orders vs. later
// kernel launches (stream ordering handles inter-kernel visibility).
__device__ __forceinline__ void gatomic_add_f32(float* p, float v) {
    asm volatile("global_atomic_add_f32 %0, %1, off scope:SCOPE_DEV"
                 :: "v"(p), "v"(v) : "memory");
}

// ---------------------------------------------------------------------------
// Degree kernels: deg starts at 1.0 (self loop), +1 per incoming edge, then
// dis = rsqrt(deg).
// ---------------------------------------------------------------------------
__global__ __launch_bounds__(256) void init_deg_kernel(float* __restrict__ deg) {
    int i = blockIdx.x * 256 + threadIdx.x;
    if (i < NNODES) deg[i] = 1.0f;
}

__global__ __launch_bounds__(256) void accum_deg_kernel(const int* __restrict__ dst,
                                                        float* __restrict__ deg) {
    int e = blockIdx.x * 256 + threadIdx.x;
    if (e < NEDGES) gatomic_add_f32(&deg[dst[e]], 1.0f);
}

__global__ __launch_bounds__(256) void rsqrt_deg_kernel(float* __restrict__ deg) {
    int i = blockIdx.x * 256 + threadIdx.x;
    if (i < NNODES) deg[i] = rsqrtf(deg[i]);   // deg >= 1 always
}

// ---------------------------------------------------------------------------
// GEMM via V_WMMA_F32_16X16X4_F32. One wave computes one 16x16 tile of
// out[M x 64] = A[M x K] @ W[K x 64] (+ optional bias). K in {32, 64}.
// A-frag layout (ISA 7.12.2, 32-bit A 16x4): lanes 0-15 -> K=0,1 in v0,v1;
// lanes 16-31 -> K=2,3. B-frag mirrored (row striped across lanes).
// C/D: VGPR v holds M=v (lanes 0-15) and M=v+8 (lanes 16-31).
// ---------------------------------------------------------------------------
__global__ __launch_bounds__(256) void gemm_wmma_kernel(const float* __restrict__ A,
                                                        const float* __restrict__ W,
                                                        const float* __restrict__ bias,
                                                        float* __restrict__ out,
                                                        int M, int K) {
    const int wave = threadIdx.x >> 5;
    const int lane = threadIdx.x & 31;
    const int tile = blockIdx.x * 8 + wave;
    const int tiles_n = HDIM / 16;                  // 4
    const int mt = tile / tiles_n;
    const int nt = tile - mt * tiles_n;
    if (mt * 16 >= M) return;                       // wave-uniform

    const int half = lane >> 4;                     // 0: K=0,1   1: K=2,3
    const int l    = lane & 15;
    const float* arow = A + (size_t)(mt * 16 + l) * K;
    const int bcol = nt * 16 + l;

    v8f c = {};
    for (int kk = 0; kk < K; kk += 4) {
        v2f a = *(const v2f*)(arow + kk + 2 * half);
        v2f b;
        b.x = W[(kk + 2 * half + 0) * HDIM + bcol];
        b.y = W[(kk + 2 * half + 1) * HDIM + bcol];
        // 8 args: (neg_a, A, neg_b, B, c_mod, C, reuse_a, reuse_b)
        c = __builtin_amdgcn_wmma_f32_16x16x4_f32(false, a, false, b,
                                                  (short)0, c, false, false);
    }

    const float bv = bias ? bias[bcol] : 0.0f;
    float* orow = out + (size_t)(mt * 16 + half * 8) * HDIM + bcol;
#pragma unroll
    for (int v = 0; v < 8; ++v) orow[v * HDIM] = c[v] + bv;
}

// ---------------------------------------------------------------------------
// Edge scatter: one wave per edge. norm = dis[src]*dis[dst] on the fly.
// Lanes read h2[src] as float2 (contiguous 256B row) and atomically add the
// scaled message into agg[dst].
// ---------------------------------------------------------------------------
__global__ __launch_bounds__(256) void scatter_kernel(const int* __restrict__ src,
                                                      const int* __restrict__ dst,
                                                      const float* __restrict__ dis,
                                                      const float* __restrict__ h2,
                                                      float* __restrict__ agg) {
    const int e = blockIdx.x * 8 + (threadIdx.x >> 5);
    const int lane = threadIdx.x & 31;
    if (e >= NEDGES) return;
    const int s = src[e];
    const int d = dst[e];
    const float nrm = dis[s] * dis[d];
    v2f v = *(const v2f*)(h2 + s * HDIM + lane * 2);
    float* p = agg + d * HDIM + lane * 2;
    gatomic_add_f32(p + 0, v.x * nrm);
    gatomic_add_f32(p + 1, v.y * nrm);
}

// ---------------------------------------------------------------------------
// Self-loop (h2[i] * dis[i]^2) + bias + ReLU epilogue. agg may alias hout.
// ---------------------------------------------------------------------------
__global__ __launch_bounds__(256) void finish_layer_kernel(const float* __restrict__ agg,
                                                           const float* __restrict__ h2,
                                                           const float* __restrict__ dis,
                                                           const float* __restrict__ bias,
                                                           float* __restrict__ hout) {
    const int i = blockIdx.x * 256 + threadIdx.x;          // element over N*H
    if (i >= NNODES * HDIM) return;
    const int node = i >> 6;
    const int j = i & 63;
    const float ds = dis[node];
    const float val = agg[i] + h2[i] * ds * ds + bias[j];
    hout[i] = fmaxf(val, 0.0f);
}

// ---------------------------------------------------------------------------
// Column sum over nodes (for mean pooling): LDS partial reduction, then 64
// atomics per block into gsum[64].
// ---------------------------------------------------------------------------
__global__ __launch_bounds__(256) void colsum_kernel(const float* __restrict__ h,
                                                     float* __restrict__ gsum) {
    __shared__ float smem[256];
    const int j  = threadIdx.x & 63;
    const int r0 = blockIdx.x * 4 + (threadIdx.x >> 6);
    float acc = 0.0f;
    for (int r = r0; r < NNODES; r += gridDim.x * 4)
        acc += h[(size_t)r * HDIM + j];
    smem[threadIdx.x] = acc;
    __syncthreads();
    if (threadIdx.x < 64) {
        float s = smem[threadIdx.x] + smem[threadIdx.x + 64] +
                  smem[threadIdx.x + 128] + smem[threadIdx.x + 192];
        gatomic_add_f32(&gsum[j], s);
    }
}

// ---------------------------------------------------------------------------
// Head: mean pool, global-feature MLP, concat, fc1+ReLU, fc2. 64 threads.
// ---------------------------------------------------------------------------
__global__ __launch_bounds__(64) void head_kernel(const float* __restrict__ gsum,
                                                  const float* __restrict__ glob,
                                                  const float* __restrict__ w_glob,
                                                  const float* __restrict__ b_glob,
                                                  const float* __restrict__ w_fc1,
                                                  const float* __restrict__ b_fc1,
                                                  const float* __restrict__ w_fc2,
                                                  const float* __restrict__ b_fc2,
                                                  float* __restrict__ out) {
    __shared__ float xc[2 * HDIM];
    __shared__ float hid[HDIM];
    const int j = threadIdx.x;

    xc[j] = gsum[j] * (1.0f / (float)NNODES);              // x_graph
    float g = b_glob[j];
#pragma unroll
    for (int k = 0; k < FGLOB; ++k) g += glob[k] * w_glob[k * HDIM + j];
    xc[HDIM + j] = fmaxf(g, 0.0f);                          // x_global
    __syncthreads();

    float a = b_fc1[j];
    for (int k = 0; k < 2 * HDIM; ++k) a += xc[k] * w_fc1[k * HDIM + j];
    hid[j] = fmaxf(a, 0.0f);
    __syncthreads();

    float o = b_fc2[j];
    for (int k = 0; k < HDIM; ++k) o += hid[k] * w_fc2[k * OUTDIM + j];
    out[j] = o;
}

// ---------------------------------------------------------------------------
extern "C" void kernel_launch(void* const* d_in, const int* in_sizes, int n_in,
                              void* d_out, int out_size, void* d_ws, size_t ws_size,
                              hipStream_t stream) {
    const float* x      = (const float*)d_in[0];
    const int*   eidx   = (const int*)d_in[1];     // [2, E] int32
    const float* glob   = (const float*)d_in[2];
    const float* w_node = (const float*)d_in[3];
    const float* b_node = (const float*)d_in[4];
    const float* w_glob = (const float*)d_in[5];
    const float* b_glob = (const float*)d_in[6];
    const float* w_c[3] = {(const float*)d_in[7], (const float*)d_in[9],  (const float*)d_in[11]};
    const float* b_c[3] = {(const float*)d_in[8], (const float*)d_in[10], (const float*)d_in[12]};
    const float* w_fc1  = (const float*)d_in[13];
    const float* b_fc1  = (const float*)d_in[14];
    const float* w_fc2  = (const float*)d_in[15];
    const float* b_fc2  = (const float*)d_in[16];
    float* out = (float*)d_out;

    const int* srcp = eidx;
    const int* dstp = eidx + NEDGES;

    // Workspace layout (~51.6 MB): h, h2 [N*H floats each], dis [N], gsum [64]
    float* h    = (float*)d_ws;
    float* h2   = h   + (size_t)NNODES * HDIM;
    float* dis  = h2  + (size_t)NNODES * HDIM;
    float* gsum = dis + NNODES;

    // 1) degrees -> dis = deg^{-1/2}
    init_deg_kernel<<<(NNODES + 255) / 256, 256, 0, stream>>>(dis);
    accum_deg_kernel<<<(NEDGES + 255) / 256, 256, 0, stream>>>(dstp, dis);
    rsqrt_deg_kernel<<<(NNODES + 255) / 256, 256, 0, stream>>>(dis);

    // 2) node embedding: h = x @ w_node + b_node  (WMMA, K=32)
    const int gemm_blocks = (NNODES / 16) * (HDIM / 16) / 8;   // 3125
    gemm_wmma_kernel<<<gemm_blocks, 256, 0, stream>>>(x, w_node, b_node, h, NNODES, FNODE);

    // 3) three GCN layers
    for (int l = 0; l < 3; ++l) {
        gemm_wmma_kernel<<<gemm_blocks, 256, 0, stream>>>(h, w_c[l], nullptr, h2, NNODES, HDIM);
        // h is dead after the GEMM -> reuse it as the scatter accumulator
        hipMemsetAsync(h, 0, (size_t)NNODES * HDIM * sizeof(float), stream);
        scatter_kernel<<<(NEDGES + 7) / 8, 256, 0, stream>>>(srcp, dstp, dis, h2, h);
        finish_layer_kernel<<<(NNODES * HDIM + 255) / 256, 256, 0, stream>>>(h, h2, dis, b_c[l], h);
    }

    // 4) mean pool + head
    hipMemsetAsync(gsum, 0, HDIM * sizeof(float), stream);
    colsum_kernel<<<1024, 256, 0, stream>>>(h, gsum);
    head_kernel<<<1, 64, 0, stream>>>(gsum, glob, w_glob, b_glob,
                                      w_fc1, b_fc1, w_fc2, b_fc2, out);
}